// RGCN_7370163880612
// MI455X (gfx1250) — compile-verified
//
#include <hip/hip_runtime.h>

// Problem constants (fixed by the reference)
#define NN      100000
#define NE      600000
#define NREL    8
#define NBASE   4
#define KTOT    640            // 128 (root part) + 4*128 (basis parts)
#define LDSTR   644            // 640 + 4 pad: (644 % 64) == 4 -> conflict-free A reads

typedef __attribute__((ext_vector_type(2))) float v2f;
typedef __attribute__((ext_vector_type(8))) float v8f;
typedef int v4i_ __attribute__((vector_size(16)));   // matches async-LDS builtin param

#define AS1 __attribute__((address_space(1)))
#define AS3 __attribute__((address_space(3)))

#if defined(__has_builtin)
#if __has_builtin(__builtin_amdgcn_global_load_async_to_lds_b128) && \
    __has_builtin(__builtin_amdgcn_s_wait_asynccnt)
#define USE_ASYNC_LDS 1
#endif
#endif

// ---------------------------------------------------------------------------
// Pass 0: build fused, transposed weights  Wt[Cpad][640]:
//   Wt[n][k] = (k < 128 ? root[k][n] : basis[(k-128)/128][(k-128)%128][n]),
//   rows n >= C are zero (ragged-tile padding -> no predication in GEMM).
// Makes every WMMA B-fragment pair a contiguous 8-byte load.
// ---------------------------------------------------------------------------
template <int C, int CPAD>
__global__ __launch_bounds__(256)
void build_wt(const float* __restrict__ root, const float* __restrict__ basis,
              float* __restrict__ Wt) {
  int i = blockIdx.x * 256 + threadIdx.x;
  if (i >= CPAD * KTOT) return;
  int n = i / KTOT;
  int k = i % KTOT;
  float v = 0.0f;
  if (n < C) v = (k < 128) ? root[k * C + n] : basis[(size_t)(k - 128) * C + n];
  Wt[(size_t)n * KTOT + k] = v;
}

// ---------------------------------------------------------------------------
// Pass 1: per-(node, relation) in-degree counts (shared by both layers).
// ---------------------------------------------------------------------------
__global__ __launch_bounds__(256)
void rgcn_count(const int* __restrict__ dst, const int* __restrict__ et,
                int* __restrict__ cnt, int E) {
  int e = blockIdx.x * blockDim.x + threadIdx.x;
  if (e < E) atomicAdd(&cnt[dst[e] * NREL + et[e]], 1);
}

// ---------------------------------------------------------------------------
// Pass 2: aggregate-first edge scatter.
//   zb[d, b, :] += (comp[t,b] / max(cnt[d,t],1)) * X[s, :]
// One wave32 per edge; each lane owns 4 contiguous channels.
// ---------------------------------------------------------------------------
__global__ __launch_bounds__(256)
void rgcn_scatter(const float* __restrict__ X, const int* __restrict__ src,
                  const int* __restrict__ dst, const int* __restrict__ et,
                  const float* __restrict__ comp, const int* __restrict__ cnt,
                  float* __restrict__ zb, int E) {
  int e    = blockIdx.x * 8 + (threadIdx.x >> 5);
  int lane = threadIdx.x & 31;
  if (e >= E) return;
  int s = src[e], d = dst[e], t = et[e];
  float inv = 1.0f / fmaxf((float)cnt[d * NREL + t], 1.0f);
  float c0 = comp[t * NBASE + 0] * inv;
  float c1 = comp[t * NBASE + 1] * inv;
  float c2 = comp[t * NBASE + 2] * inv;
  float c3 = comp[t * NBASE + 3] * inv;
  float4 v = *(const float4*)(X + (size_t)s * 128 + lane * 4);
  float* zr = zb + (size_t)d * (NBASE * 128) + lane * 4;
  atomicAdd(zr + 0 * 128 + 0, c0 * v.x);
  atomicAdd(zr + 0 * 128 + 1, c0 * v.y);
  atomicAdd(zr + 0 * 128 + 2, c0 * v.z);
  atomicAdd(zr + 0 * 128 + 3, c0 * v.w);
  atomicAdd(zr + 1 * 128 + 0, c1 * v.x);
  atomicAdd(zr + 1 * 128 + 1, c1 * v.y);
  atomicAdd(zr + 1 * 128 + 2, c1 * v.z);
  atomicAdd(zr + 1 * 128 + 3, c1 * v.w);
  atomicAdd(zr + 2 * 128 + 0, c2 * v.x);
  atomicAdd(zr + 2 * 128 + 1, c2 * v.y);
  atomicAdd(zr + 2 * 128 + 2, c2 * v.z);
  atomicAdd(zr + 2 * 128 + 3, c2 * v.w);
  atomicAdd(zr + 3 * 128 + 0, c3 * v.x);
  atomicAdd(zr + 3 * 128 + 1, c3 * v.y);
  atomicAdd(zr + 3 * 128 + 2, c3 * v.z);
  atomicAdd(zr + 3 * 128 + 3, c3 * v.w);
}

// ---------------------------------------------------------------------------
// Pass 3: fused node GEMM via fp32 WMMA (bit-exact precision vs reference).
//   Out[m, n] = act( [X | zb][m, 0:640] @ Wt[n, 0:640]^T + bias[n] )
// Block = 256 threads = 8 waves; block owns 16 rows, A staged once in LDS
// (async direct-to-LDS when available); wave w owns cols [16w, 16w+16).
// Inner loop: ds_load_b64 (A) + global_load_b64 (B, immediate offsets)
// + v_wmma_f32_16x16x4_f32, 160 steps, accumulator chained in registers.
// ---------------------------------------------------------------------------
template <int C, int CPAD, bool RELU>
__global__ __launch_bounds__(256)
void rgcn_gemm(const float* __restrict__ X, const float* __restrict__ Z,
               const float* __restrict__ Wt, const float* __restrict__ bias,
               float* __restrict__ Out) {
  __shared__ float Atile[16 * LDSTR];

  int tid  = threadIdx.x;
  int lane = tid & 31;
  int wid  = tid >> 5;
  int m0   = blockIdx.x * 16;

  // Stage the 16 x 640 A tile: rows = [x | zb] concatenated, float4 granules.
  for (int i = tid; i < 16 * 160; i += 256) {
    int r  = i / 160;
    int c4 = i % 160;
    const float* g = (c4 < 32) ? (X + (size_t)(m0 + r) * 128 + c4 * 4)
                               : (Z + (size_t)(m0 + r) * 512 + (c4 - 32) * 4);
    float* l = &Atile[r * LDSTR + c4 * 4];
#ifdef USE_ASYNC_LDS
    __builtin_amdgcn_global_load_async_to_lds_b128((AS1 v4i_*)g, (AS3 v4i_*)l,
                                                   0, 0);
#else
    *(float4*)l = *(const float4*)g;
#endif
  }
#ifdef USE_ASYNC_LDS
  __builtin_amdgcn_s_wait_asynccnt(0);
#endif
  __syncthreads();

  int n0 = wid * 16;
  if (n0 >= CPAD) return;            // wave-uniform: EXEC all-ones through WMMA

  int ncol  = n0 + (lane & 15);
  int khalf = (lane >> 4) * 2;       // K sub-lane per ISA 32-bit A/B layout
  const float* arow = &Atile[(lane & 15) * LDSTR + khalf];
  const float* wrow = Wt + (size_t)ncol * KTOT + khalf;   // padded rows are 0

  v8f acc = {};
  #pragma unroll 8
  for (int kk = 0; kk < KTOT; kk += 4) {
    v2f a = *(const v2f*)(arow + kk);   // LDS, imm offset <= 2556
    v2f b = *(const v2f*)(wrow + kk);   // global, imm offset <= 2556
    acc = __builtin_amdgcn_wmma_f32_16x16x4_f32(false, a, false, b,
                                                (short)0, acc, false, false);
  }

  // C/D layout: VGPR j -> row m0 + 8*(lane>>4) + j, col = n0 + (lane&15)
  if (ncol < C) {
    float bi = bias[ncol];
    int   hi = (lane >> 4) * 8;
    #pragma unroll
    for (int j = 0; j < 8; ++j) {
      float v = acc[j] + bi;
      if (RELU) v = fmaxf(v, 0.0f);
      Out[(size_t)(m0 + hi + j) * C + ncol] = v;
    }
  }
}

// ---------------------------------------------------------------------------
extern "C" void kernel_launch(void* const* d_in, const int* in_sizes, int n_in,
                              void* d_out, int out_size, void* d_ws, size_t ws_size,
                              hipStream_t stream) {
  const float* x      = (const float*)d_in[0];
  const int*   eidx   = (const int*)d_in[1];   // [2, E]
  const int*   etype  = (const int*)d_in[2];   // [E]
  const float* comp1  = (const float*)d_in[3];
  const float* basis1 = (const float*)d_in[4];
  const float* root1  = (const float*)d_in[5];
  const float* bias1  = (const float*)d_in[6];
  const float* comp2  = (const float*)d_in[7];
  const float* basis2 = (const float*)d_in[8];
  const float* root2  = (const float*)d_in[9];
  const float* bias2  = (const float*)d_in[10];
  float*       out    = (float*)d_out;

  const int* srcI = eidx;
  const int* dstI = eidx + NE;

  // Workspace: zb (204.8MB) | h (51.2MB) | cnt (3.2MB) | Wt1 (320KB) | Wt2 (120KB)
  size_t ZB_BYTES  = (size_t)NN * NBASE * 128 * sizeof(float);
  size_t H_BYTES   = (size_t)NN * 128 * sizeof(float);
  size_t CNT_BYTES = (size_t)NN * NREL * sizeof(int);
  size_t WT1_BYTES = (size_t)128 * KTOT * sizeof(float);
  float* zb  = (float*)d_ws;
  float* h   = (float*)((char*)d_ws + ZB_BYTES);
  int*   cnt = (int*)((char*)d_ws + ZB_BYTES + H_BYTES);
  float* wt1 = (float*)((char*)d_ws + ZB_BYTES + H_BYTES + CNT_BYTES);
  float* wt2 = (float*)((char*)d_ws + ZB_BYTES + H_BYTES + CNT_BYTES + WT1_BYTES);

  dim3 gGemm(NN / 16, 1, 1);           // 6250 row tiles, exact

  // Pass 0: fused transposed weights (tiny)
  build_wt<128, 128><<<(128 * KTOT + 255) / 256, 256, 0, stream>>>(root1, basis1, wt1);
  build_wt<40, 48><<<(48 * KTOT + 255) / 256, 256, 0, stream>>>(root2, basis2, wt2);

  // Pass 1: shared count pass (edge types identical across layers)
  (void)hipMemsetAsync(cnt, 0, CNT_BYTES, stream);
  rgcn_count<<<(NE + 255) / 256, 256, 0, stream>>>(dstI, etype, cnt, NE);

  // ---- Layer 1: x -> h (ReLU) ----
  (void)hipMemsetAsync(zb, 0, ZB_BYTES, stream);
  rgcn_scatter<<<(NE + 7) / 8, 256, 0, stream>>>(x, srcI, dstI, etype,
                                                 comp1, cnt, zb, NE);
  rgcn_gemm<128, 128, true><<<gGemm, 256, 0, stream>>>(x, zb, wt1, bias1, h);

  // ---- Layer 2: h -> out ----
  (void)hipMemsetAsync(zb, 0, ZB_BYTES, stream);
  rgcn_scatter<<<(NE + 7) / 8, 256, 0, stream>>>(h, srcI, dstI, etype,
                                                 comp2, cnt, zb, NE);
  rgcn_gemm<40, 48, false><<<gGemm, 256, 0, stream>>>(h, zb, wt2, bias2, out);
}